// PerPixelConv_2534030705069
// MI455X (gfx1250) — compile-verified
//
#include <hip/hip_runtime.h>
#include <stdint.h>

// ---------------------------------------------------------------------------
// Per-pixel dynamic 3x3 conv (B=8, C=3, H=720, W=1280, f32).
// HBM-bound (~0.9 FLOP/B at 23.3 TB/s -> ~20 us floor): optimize data movement.
//  - Input halo rows staged to LDS via TDM (tensor_load_to_lds, TENSORcnt);
//    bottom boundary zero-filled by TDM OOB semantics, top row zeroed manually.
//  - Kernel planes (265 MB, touched once) streamed with NT b128 loads.
//  - Outputs (88.5 MB, touched once) stored with NT b128 stores.
//  - L2 (192 MB) is reserved for the input (88.5 MB), the only reused array.
// ---------------------------------------------------------------------------

constexpr int B_ = 8, C_ = 3, H_ = 720, W_ = 1280;
constexpr int ROWS  = 2;           // output rows per block
constexpr int SROWS = ROWS + 2;    // staged input rows (halo) per channel
constexpr int TPB   = 320;         // 10 wave32s; 320*4 px = 1280 = W
// LDS: 3 ch * 4 rows * 1280 f32 = 61440 B

typedef unsigned int u32x4 __attribute__((ext_vector_type(4)));
typedef int          i32x8 __attribute__((ext_vector_type(8)));
typedef int          i32x4 __attribute__((ext_vector_type(4)));
typedef float        f32x4 __attribute__((ext_vector_type(4)));  // native clang vec
                                                                 // (NT builtins need it)

// Issue one TDM 2-D tile load: global (tensor_d0 x tensor_d1, row stride
// d0_stride elems of 4B) -> LDS (contiguous tile_d0 x tile_d1).
// Rows with y >= tensor_d1 are zero-filled by TDM OOB semantics.
__device__ __forceinline__ void tdm_load_2d(unsigned ldsByteAddr, const void* gptr,
                                            unsigned tensor_d0, unsigned tensor_d1,
                                            unsigned tile_d0, unsigned tile_d1,
                                            unsigned d0_stride) {
  const uint64_t ga = (uint64_t)(uintptr_t)gptr;

  u32x4 g0;
  g0[0] = 1u;                                               // count=1, user mode
  g0[1] = ldsByteAddr;                                      // lds_addr [63:32]
  g0[2] = (unsigned)(ga & 0xFFFFFFFFu);                     // global_addr lo
  g0[3] = (unsigned)((ga >> 32) & 0x01FFFFFFu) | (2u << 30);// addr[56:32] | type=2

  i32x8 g1;
  g1[0] = (int)(2u << 16);                                  // data_size = 4 bytes
  g1[1] = (int)((tensor_d0 & 0xFFFFu) << 16);               // tensor_dim0 lo16
  g1[2] = (int)(((tensor_d0 >> 16) & 0xFFFFu)
              | ((tensor_d1 & 0xFFFFu) << 16));             // dim0 hi | dim1 lo
  g1[3] = (int)(((tensor_d1 >> 16) & 0xFFFFu)
              | ((tile_d0 & 0xFFFFu) << 16));               // dim1 hi | tile_dim0
  g1[4] = (int)(tile_d1 & 0xFFFFu);                         // tile_dim1 | tile_dim2=0
  g1[5] = (int)d0_stride;                                   // tensor_dim0_stride lo32
  g1[6] = 0;                                                // stride hi | dim1_stride lo
  g1[7] = 0;

  i32x4 g2 = {0, 0, 0, 0};                                  // 2-D: no dim2/3
  i32x4 g3 = {0, 0, 0, 0};

#if defined(__clang_major__) && __clang_major__ >= 23
  i32x8 gz = {0, 0, 0, 0, 0, 0, 0, 0};
  __builtin_amdgcn_tensor_load_to_lds(g0, g1, g2, g3, gz, 0);
#else
  __builtin_amdgcn_tensor_load_to_lds(g0, g1, g2, g3, 0);
#endif
}

__global__ __launch_bounds__(TPB) void PerPixelConv_kernel(
    const float* __restrict__ in,    // [B,3,H,W]
    const float* __restrict__ ker,   // [B,9,H,W]
    float* __restrict__ out) {       // [B,3,H,W]
  __shared__ float smem[C_ * SROWS * W_];

  const int tid = threadIdx.x;
  const int b   = blockIdx.y;
  const int h0  = blockIdx.x * ROWS;

  const int r_start = (h0 == 0) ? 0 : (h0 - 1);
  const int row_off = (h0 == 0) ? 1 : 0;       // skip LDS row 0, zero it manually
  const unsigned ldsBase = (unsigned)(uintptr_t)&smem[0];  // flat addr[31:0] = LDS offset

  // Top image boundary: zero pad row (row 0 of each channel's stage buffer).
  // Division-free indexing; runs only for the 8 h0==0 blocks.
  if (h0 == 0) {
#pragma unroll
    for (int c = 0; c < C_; ++c)
      for (int x = tid; x < W_; x += TPB)
        smem[(c * SROWS) * W_ + x] = 0.0f;
  }

  // Wave 0 issues one TDM tile load per channel, then drains TENSORcnt.
  // Bottom boundary handled by TDM OOB zero-fill (tensor_dim1 = H - r_start).
  if (tid < 32) {
#pragma unroll
    for (int c = 0; c < C_; ++c) {
      const float* src = in + (((size_t)b * C_ + c) * H_ + r_start) * W_;
      const unsigned lds = ldsBase + (unsigned)(((c * SROWS + row_off) * W_) * 4);
      tdm_load_2d(lds, src,
                  (unsigned)W_, (unsigned)(H_ - r_start),
                  (unsigned)W_, (unsigned)(SROWS - row_off),
                  (unsigned)W_);
    }
    __builtin_amdgcn_s_wait_tensorcnt(0);
  }
  __syncthreads();

  const int w0 = tid * 4;  // 4 consecutive pixels per thread

  // Rolling 3x6 register window per channel (cols w0-1 .. w0+4).
  float wreg[C_][3][6];
#pragma unroll
  for (int c = 0; c < C_; ++c)
#pragma unroll
    for (int j = 0; j < 3; ++j) {
      const float* r = &smem[(c * SROWS + j) * W_];
      wreg[c][j][0] = (w0 > 0) ? r[w0 - 1] : 0.0f;
      wreg[c][j][1] = r[w0 + 0];
      wreg[c][j][2] = r[w0 + 1];
      wreg[c][j][3] = r[w0 + 2];
      wreg[c][j][4] = r[w0 + 3];
      wreg[c][j][5] = (w0 + 4 < W_) ? r[w0 + 4] : 0.0f;
    }

  const float* kerB = ker + (size_t)b * 9 * H_ * W_;

#pragma unroll
  for (int rr = 0; rr < ROWS; ++rr) {
    const int h = h0 + rr;

    // 9 per-pixel weight vectors: single-use bulk traffic -> NT b128 loads
    // (TH=NT keeps L2 free for the reused input array).
    f32x4 kf[9];
#pragma unroll
    for (int k = 0; k < 9; ++k)
      kf[k] = __builtin_nontemporal_load(
          reinterpret_cast<const f32x4*>(kerB + ((size_t)k * H_ + h) * W_ + w0));

#pragma unroll
    for (int c = 0; c < C_; ++c) {
      float ax = 0.f, ay = 0.f, az = 0.f, aw = 0.f;
#pragma unroll
      for (int dy = 0; dy < 3; ++dy)
#pragma unroll
        for (int dx = 0; dx < 3; ++dx) {
          const f32x4 kv = kf[dy * 3 + dx];
          ax = fmaf(wreg[c][dy][0 + dx], kv.x, ax);
          ay = fmaf(wreg[c][dy][1 + dx], kv.y, ay);
          az = fmaf(wreg[c][dy][2 + dx], kv.z, az);
          aw = fmaf(wreg[c][dy][3 + dx], kv.w, aw);
        }
      // Single-use output -> NT b128 store.
      f32x4 o;
      o.x = ax; o.y = ay; o.z = az; o.w = aw;
      __builtin_nontemporal_store(
          o, reinterpret_cast<f32x4*>(out + (((size_t)b * C_ + c) * H_ + h) * W_ + w0));
    }

    // Slide the vertical window: one new LDS row per channel.
    if (rr + 1 < ROWS) {
#pragma unroll
      for (int c = 0; c < C_; ++c) {
#pragma unroll
        for (int j = 0; j < 2; ++j)
#pragma unroll
          for (int x = 0; x < 6; ++x) wreg[c][j][x] = wreg[c][j + 1][x];
        const float* r = &smem[(c * SROWS + rr + 3) * W_];
        wreg[c][2][0] = (w0 > 0) ? r[w0 - 1] : 0.0f;
        wreg[c][2][1] = r[w0 + 0];
        wreg[c][2][2] = r[w0 + 1];
        wreg[c][2][3] = r[w0 + 2];
        wreg[c][2][4] = r[w0 + 3];
        wreg[c][2][5] = (w0 + 4 < W_) ? r[w0 + 4] : 0.0f;
      }
    }
  }
}

extern "C" void kernel_launch(void* const* d_in, const int* in_sizes, int n_in,
                              void* d_out, int out_size, void* d_ws, size_t ws_size,
                              hipStream_t stream) {
  (void)in_sizes; (void)n_in; (void)out_size; (void)d_ws; (void)ws_size;
  const float* in  = (const float*)d_in[0];   // [8,3,720,1280] f32
  const float* ker = (const float*)d_in[1];   // [8,9,720,1280] f32
  float* out = (float*)d_out;                 // [8,3,720,1280] f32

  dim3 grid(H_ / ROWS, B_);                   // (360, 8) blocks
  PerPixelConv_kernel<<<grid, TPB, 0, stream>>>(in, ker, out);
}